// SSNFeatsTransformAdaptive_86552180949441
// MI455X (gfx1250) — compile-verified
//
#include <hip/hip_runtime.h>
#include <math.h>

// Problem constants (from reference)
constexpr int Bb = 2, Cc = 32, Hh = 192, Ww = 256;
constexpr int Jh = 12, Jw = 16, Jj = Jh * Jw;       // 192 superpixels
constexpr int Nn = Hh * Ww;                          // 49152 pixels
constexpr int BDIM = 256;
constexpr float BIGF = 1e16f;

// Workspace layout (float offsets)
constexpr int SPIX_OFF = 0;             // B*C*J = 12288
constexpr int NUM_OFF  = 12288;         // 12288
constexpr int DEN_OFF  = 24576;         // 384 (contiguous after NUM)
constexpr int FMJ_OFF  = 24960;         // 12288

typedef __attribute__((ext_vector_type(2))) float v2f;
typedef __attribute__((ext_vector_type(8))) float v8f;

// ---------------- zero helpers ----------------
__global__ void zero_f4_k(float4* __restrict__ p, int n4) {
    int i = blockIdx.x * BDIM + threadIdx.x;
    if (i < n4) p[i] = make_float4(0.f, 0.f, 0.f, 0.f);
}
__global__ void zero_f_k(float* __restrict__ p, int n) {
    int i = blockIdx.x * BDIM + threadIdx.x;
    if (i < n) p[i] = 0.f;
}

// ---------------- spix init: 16x16 block mean ----------------
// grid = B*J blocks, 256 threads = 8 waves; wave w -> channels 4w..4w+3
__global__ void init_spix_k(const float* __restrict__ feats, float* __restrict__ spix) {
    int blk = blockIdx.x;
    int b = blk / Jj, j = blk % Jj;
    int jy = j / Jw, jx = j % Jw;
    int wave = threadIdx.x >> 5, lane = threadIdx.x & 31;
    int y0 = jy * 16, x0 = jx * 16;
    for (int ci = 0; ci < 4; ++ci) {
        int c = wave * 4 + ci;
        const float* base = feats + (size_t)(b * Cc + c) * Nn;
        float s = 0.f;
        #pragma unroll
        for (int i = 0; i < 8; ++i) {
            int p = lane + i * 32;
            int y = y0 + (p >> 4), x = x0 + (p & 15);
            s += base[y * Ww + x];
        }
        #pragma unroll
        for (int off = 16; off; off >>= 1) s += __shfl_xor(s, off, 32);
        if (lane == 0) spix[(b * Cc + c) * Jj + j] = s * (1.f / 256.f);
    }
}

// ---------------- one SSN iteration ----------------
// grid = B*J blocks (one per cell), 256 threads (one per pixel of the cell).
// Per-cell numerator reduction num_cell[k,c] = sum_p aff[k,p]*pix[p,c] is a
// (16x256)x(256x32) GEMM done with V_WMMA_F32_16X16X4_F32 by the 8 waves
// (2 channel tiles x 4-way K split), instead of same-address LDS atomics.
__global__ void ssn_iter_k(const float* __restrict__ feats,
                           const float* __restrict__ spix,
                           float* __restrict__ num, float* __restrict__ den,
                           float* __restrict__ gamma, int last) {
    __shared__ float s_sp[9][32];      // neighbor centroids
    __shared__ float s_pix[256][33];   // B matrix: pixel x channel (padded pitch 33)
    __shared__ float s_aff[16][257];   // A matrix: k x pixel (padded pitch 257); rows 9..15 unused
    __shared__ float s_acc[16][32];    // GEMM result merge
    __shared__ float s_den[9];
    int blk = blockIdx.x;
    int b = blk / Jj, j = blk % Jj;
    int jy = j / Jw, jx = j % Jw;
    int tid = threadIdx.x;
    int lane = tid & 31;

    for (int idx = tid; idx < 288; idx += BDIM) {
        int k = idx >> 5, c = idx & 31;
        int dy = k / 3 - 1, dx = k % 3 - 1;
        int ny = jy + dy, nx = jx + dx;
        bool valid = ((unsigned)ny < (unsigned)Jh) && ((unsigned)nx < (unsigned)Jw);
        int jn = ny * Jw + nx;
        s_sp[k][c] = valid ? spix[(b * Cc + c) * Jj + jn] : 0.f;
    }
    for (int idx = tid; idx < 512; idx += BDIM) ((float*)s_acc)[idx] = 0.f;
    if (tid < 9) s_den[tid] = 0.f;
    __syncthreads();

    int ty = tid >> 4, tx = tid & 15;
    int y = jy * 16 + ty, x = jx * 16 + tx;
    int n = y * Ww + x;

    float f[32];
    #pragma unroll
    for (int c = 0; c < 32; ++c) f[c] = feats[(size_t)(b * Cc + c) * Nn + n];

    float d[9];
    float dmin = BIGF;
    #pragma unroll
    for (int k = 0; k < 9; ++k) {
        int dy = k / 3 - 1, dx = k % 3 - 1;
        int ny = jy + dy, nx = jx + dx;
        bool valid = ((unsigned)ny < (unsigned)Jh) && ((unsigned)nx < (unsigned)Jw);
        float s = 0.f;
        #pragma unroll
        for (int c = 0; c < 32; ++c) { float t = f[c] - s_sp[k][c]; s += t * t; }
        d[k] = valid ? s : BIGF;
        dmin = fminf(dmin, d[k]);
    }
    float aff[9], esum = 0.f;
    #pragma unroll
    for (int k = 0; k < 9; ++k) { aff[k] = __expf(dmin - d[k]); esum += aff[k]; }
    float inv = 1.f / esum;
    #pragma unroll
    for (int k = 0; k < 9; ++k) aff[k] *= inv;

    // Stage GEMM operands (invalid neighbors have aff == 0 exactly)
    #pragma unroll
    for (int c = 0; c < 32; ++c) s_pix[tid][c] = f[c];
    #pragma unroll
    for (int k = 0; k < 9; ++k) s_aff[k][tid] = aff[k];

    // den[k] = sum_p aff[k,p]: wave shuffle reduce + one LDS atomic per wave
    #pragma unroll
    for (int k = 0; k < 9; ++k) {
        float v = aff[k];
        #pragma unroll
        for (int off = 16; off; off >>= 1) v += __shfl_xor(v, off, 32);
        if (lane == 0) atomicAdd(&s_den[k], v);
    }

    // last iteration: scatter affinity into the dense gamma output
    if (last) {
        #pragma unroll
        for (int k = 0; k < 9; ++k) {
            int dy = k / 3 - 1, dx = k % 3 - 1;
            int ny = jy + dy, nx = jx + dx;
            bool valid = ((unsigned)ny < (unsigned)Jh) && ((unsigned)nx < (unsigned)Jw);
            if (valid) {
                int jn = ny * Jw + nx;
                gamma[(size_t)(b * Jj + jn) * Nn + n] = aff[k];
            }
        }
    }
    __syncthreads();

    // WMMA GEMM: wave -> (ctile = wave/4, K quarter = wave%4), 16 steps of K=4
    int wave = tid >> 5;
    int half = lane >> 4, lm = lane & 15;
    int ctile = wave >> 2, kq = wave & 3;
    int kbase = kq * 64;
    v8f acc = {};
    for (int k0 = 0; k0 < 64; k0 += 4) {
        int ka = kbase + k0 + half * 2;
        v2f a, bf;
        a.x = s_aff[lm][ka];                 // A[m=lm][K=ka..ka+1]
        a.y = s_aff[lm][ka + 1];
        bf.x = s_pix[ka][ctile * 16 + lm];   // B[K=ka..ka+1][n=c]
        bf.y = s_pix[ka + 1][ctile * 16 + lm];
        acc = __builtin_amdgcn_wmma_f32_16x16x4_f32(false, a, false, bf,
                                                    (short)0, acc, false, false);
    }
    // merge 4 K-partials per ctile (rows 9..15 are don't-care garbage)
    #pragma unroll
    for (int r = 0; r < 8; ++r) {
        int krow = r + half * 8;
        atomicAdd(&s_acc[krow][ctile * 16 + lm], acc[r]);
    }
    __syncthreads();

    // global scatter to the 9 neighbor cells
    for (int idx = tid; idx < 288; idx += BDIM) {
        int k = idx >> 5, c = idx & 31;
        int dy = k / 3 - 1, dx = k % 3 - 1;
        int ny = jy + dy, nx = jx + dx;
        bool valid = ((unsigned)ny < (unsigned)Jh) && ((unsigned)nx < (unsigned)Jw);
        if (valid) {
            int jn = ny * Jw + nx;
            atomicAdd(&num[(b * Cc + c) * Jj + jn], s_acc[k][c]);
            if (c == 0) atomicAdd(&den[b * Jj + jn], s_den[k]);
        }
    }
}

// ---------------- renorm: out = num / (den + eps) ----------------
__global__ void renorm_k(const float* __restrict__ num, const float* __restrict__ den,
                         float* __restrict__ out, float eps) {
    int i = blockIdx.x * BDIM + threadIdx.x;
    if (i >= Bb * Cc * Jj) return;
    int j = i % Jj;
    int b = i / (Cc * Jj);
    out[i] = num[i] / (den[b * Jj + j] + eps);
}

// ---------------- reconstruction: recon[b,d,n] = sum_j FMJ[b,d,j]*gamma[b,j,n] ----------------
// WMMA f32 16x16x4: M = 16 channels (mt in {0,1}), K = 192 (step 4), N = 16 pixels.
// grid = B * 2 * 384 blocks, 256 threads = 8 waves; each wave owns one 16-pixel N tile.
__global__ void recon_wmma_k(const float* __restrict__ fmj,
                             const float* __restrict__ gamma,
                             float* __restrict__ recon) {
    __shared__ float s_a[16][193];  // padded (193 % 64 == 1) to avoid bank conflicts
    int blk = blockIdx.x;
    int b = blk / 768;
    int rem = blk % 768;
    int mt = rem / 384;
    int nblk = rem % 384;
    int tid = threadIdx.x;

    for (int idx = tid; idx < 16 * 192; idx += BDIM) {
        int r = idx / 192, k = idx % 192;
        s_a[r][k] = fmj[(b * Cc + mt * 16 + r) * Jj + k];
    }
    __syncthreads();

    int wave = tid >> 5, lane = tid & 31;
    int half = lane >> 4, lm = lane & 15;
    int n0 = (nblk * 8 + wave) * 16;

    v8f acc = {};
    const float* gbase = gamma + (size_t)b * Jj * Nn + n0 + lm;
    for (int k0 = 0; k0 < 192; k0 += 4) {
        // A frag (16x4 f32): lanes 0-15 hold {K=k0,k0+1}, lanes 16-31 hold {K=k0+2,k0+3}
        int ka = k0 + half * 2;
        v2f a, bfrag;
        a.x = s_a[lm][ka];
        a.y = s_a[lm][ka + 1];
        // B frag (4x16 f32): VGPR0 = row ka, VGPR1 = row ka+1, N = lm
        const float* g = gbase + (size_t)ka * Nn;
        bfrag.x = g[0];
        bfrag.y = g[Nn];
        acc = __builtin_amdgcn_wmma_f32_16x16x4_f32(false, a, false, bfrag,
                                                    (short)0, acc, false, false);
    }
    // C/D layout: VGPR r -> M = r (lanes 0-15) / r+8 (lanes 16-31), N = lm
    #pragma unroll
    for (int r = 0; r < 8; ++r) {
        int drow = mt * 16 + r + half * 8;
        recon[(size_t)(b * Cc + drow) * Nn + n0 + lm] = acc[r];
    }
}

extern "C" void kernel_launch(void* const* d_in, const int* in_sizes, int n_in,
                              void* d_out, int out_size, void* d_ws, size_t ws_size,
                              hipStream_t stream) {
    (void)in_sizes; (void)n_in; (void)out_size; (void)ws_size;
    const float* feats = (const float*)d_in[0];
    float* out   = (float*)d_out;
    float* gamma = out;                          // B*J*N floats
    float* recon = out + (size_t)Bb * Jj * Nn;   // B*C*N floats
    float* ws   = (float*)d_ws;
    float* spix = ws + SPIX_OFF;
    float* num  = ws + NUM_OFF;
    float* den  = ws + DEN_OFF;
    float* fmj  = ws + FMJ_OFF;

    // zero the dense gamma output (only 9 entries/pixel-column get written)
    int n4 = (Bb * Jj * Nn) / 4;
    zero_f4_k<<<(n4 + BDIM - 1) / BDIM, BDIM, 0, stream>>>((float4*)gamma, n4);

    init_spix_k<<<Bb * Jj, BDIM, 0, stream>>>(feats, spix);

    for (int it = 0; it < 10; ++it) {
        int nacc = Bb * Cc * Jj + Bb * Jj;  // num + den are contiguous
        zero_f_k<<<(nacc + BDIM - 1) / BDIM, BDIM, 0, stream>>>(num, nacc);
        ssn_iter_k<<<Bb * Jj, BDIM, 0, stream>>>(feats, spix, num, den, gamma, it == 9);
        int ncj = Bb * Cc * Jj;
        if (it < 9)
            renorm_k<<<(ncj + BDIM - 1) / BDIM, BDIM, 0, stream>>>(num, den, spix, 1e-16f);
        else
            renorm_k<<<(ncj + BDIM - 1) / BDIM, BDIM, 0, stream>>>(num, den, fmj, 1e-6f);
    }

    recon_wmma_k<<<Bb * 2 * 384, BDIM, 0, stream>>>(fmj, gamma, recon);
}